// GNN_QNetwork_12678743458408
// MI455X (gfx1250) — compile-verified
//
#include <hip/hip_runtime.h>
#include <stdint.h>

typedef __attribute__((ext_vector_type(2))) float v2f;
typedef __attribute__((ext_vector_type(8))) float v8f;

#define FDIM 64   // F_IN == HID == 64
#define NACT 8

// ---------------------------------------------------------------------------
// Degree / normalization
// ---------------------------------------------------------------------------
__global__ void init_deg_kernel(float* __restrict__ deg, int n) {
  int i = blockIdx.x * blockDim.x + threadIdx.x;
  if (i < n) deg[i] = 1.0f;  // self loop
}

__global__ void count_deg_kernel(const int* __restrict__ col,
                                 float* __restrict__ deg, int e) {
  int i = blockIdx.x * blockDim.x + threadIdx.x;
  if (i < e) atomicAdd(&deg[col[i]], 1.0f);
}

__global__ void rsqrt_kernel(float* __restrict__ deg, int n) {
  int i = blockIdx.x * blockDim.x + threadIdx.x;
  if (i < n) deg[i] = rsqrtf(deg[i]);  // deg >= 1 always (self loop)
}

// ---------------------------------------------------------------------------
// OUT[N x 64] = A[N x 64] * W[64 x 64]  via V_WMMA_F32_16X16X4_F32.
// One wave per 16-row strip; 4 column tiles x 16 k-steps = 64 WMMAs/wave.
// A-frag layout (ISA 7.12.2, 32-bit A 16x4): lane m=lane&15, half=lane>>4,
//   a[v] = A[m][k0 + 2*half + v].
// B-frag (4x16, row striped across lanes): b[v] = W[k0 + 2*half + v][n].
// C/D (16x16 f32): vgpr v, M = v + 8*half, N = lane&15.
// ---------------------------------------------------------------------------
__global__ __launch_bounds__(256) void gemm64_wmma_kernel(
    const float* __restrict__ A, const float* __restrict__ W,
    float* __restrict__ OUT, int nTiles) {
  int wave = (int)((blockIdx.x * blockDim.x + threadIdx.x) >> 5);
  int lane = threadIdx.x & 31;
  if (wave >= nTiles) return;  // wave-uniform: EXEC stays all-ones

  const int r0 = wave * 16;
  const int m = lane & 15;
  const int half = lane >> 4;

  // Hold the whole 16x64 A strip in registers (32 VGPRs).
  v2f a[16];
  const float* arow = A + (size_t)(r0 + m) * FDIM + 2 * half;
#pragma unroll
  for (int ks = 0; ks < 16; ++ks) {
    a[ks].x = arow[ks * 4 + 0];
    a[ks].y = arow[ks * 4 + 1];
  }

#pragma unroll
  for (int nt = 0; nt < 4; ++nt) {
    const int n = nt * 16 + m;
    v8f c = {0.f, 0.f, 0.f, 0.f, 0.f, 0.f, 0.f, 0.f};
#pragma unroll
    for (int ks = 0; ks < 16; ++ks) {
      const int k = ks * 4 + 2 * half;
      v2f b;
      b.x = W[(size_t)k * FDIM + n];
      b.y = W[(size_t)(k + 1) * FDIM + n];
      c = __builtin_amdgcn_wmma_f32_16x16x4_f32(false, a[ks], false, b,
                                                (short)0, c, false, false);
    }
#pragma unroll
    for (int v = 0; v < 8; ++v) {
      OUT[(size_t)(r0 + v + 8 * half) * FDIM + n] = c[v];
    }
  }
}

// ---------------------------------------------------------------------------
// agg[i] = dinv[i]^2 * h[i]   (self-loop contribution; float4 per thread)
// ---------------------------------------------------------------------------
__global__ void self_loop_init_kernel(const float* __restrict__ h,
                                      const float* __restrict__ dinv,
                                      float* __restrict__ agg, int n) {
  int t = blockIdx.x * blockDim.x + threadIdx.x;  // over n*16 float4s
  if (t >= n * 16) return;
  int node = t >> 4;
  float s = dinv[node];
  s = s * s;
  float4 hv = ((const float4*)h)[t];
  float4 o;
  o.x = s * hv.x; o.y = s * hv.y; o.z = s * hv.z; o.w = s * hv.w;
  ((float4*)agg)[t] = o;
}

// ---------------------------------------------------------------------------
// Edge scatter: agg[dst] += dinv[src]*dinv[dst] * h[src]
// 16 threads per edge, float4 gather + 4x global_atomic_add_f32 each.
// ---------------------------------------------------------------------------
__global__ void scatter_edges_kernel(const int* __restrict__ ei,  // [2*E]
                                     const float* __restrict__ h,
                                     const float* __restrict__ dinv,
                                     float* __restrict__ agg, int E) {
  int t = blockIdx.x * blockDim.x + threadIdx.x;  // over E*16
  int e = t >> 4;
  if (e >= E) return;
  int sub = t & 15;
  int src = ei[e];
  int dst = ei[E + e];
  float norm = dinv[src] * dinv[dst];
  float4 hv = ((const float4*)(h + (size_t)src * FDIM))[sub];
  float* out = agg + (size_t)dst * FDIM + sub * 4;
  atomicAdd(out + 0, norm * hv.x);
  atomicAdd(out + 1, norm * hv.y);
  atomicAdd(out + 2, norm * hv.z);
  atomicAdd(out + 3, norm * hv.w);
}

// ---------------------------------------------------------------------------
// buf = relu(buf + bias)  (bias broadcast across the 64-wide feature dim)
// ---------------------------------------------------------------------------
__global__ void bias_relu_kernel(float* __restrict__ buf,
                                 const float* __restrict__ bias, int n64) {
  int t = blockIdx.x * blockDim.x + threadIdx.x;
  if (t >= n64) return;
  float v = buf[t] + bias[t & (FDIM - 1)];
  buf[t] = v > 0.0f ? v : 0.0f;
}

// ---------------------------------------------------------------------------
// out[N x 8] = h[N x 64] @ Wfc[64 x 8] + bfc   (compute-trivial head)
// ---------------------------------------------------------------------------
__global__ void fc_head_kernel(const float* __restrict__ h,
                               const float* __restrict__ Wfc,
                               const float* __restrict__ bfc,
                               float* __restrict__ out, int n) {
  int t = blockIdx.x * blockDim.x + threadIdx.x;  // over n*NACT
  if (t >= n * NACT) return;
  int node = t >> 3;
  int a = t & (NACT - 1);
  const float4* h4 = (const float4*)(h + (size_t)node * FDIM);
  float acc = bfc[a];
#pragma unroll
  for (int kk = 0; kk < FDIM / 4; ++kk) {
    float4 hv = h4[kk];
    acc += hv.x * Wfc[(kk * 4 + 0) * NACT + a];
    acc += hv.y * Wfc[(kk * 4 + 1) * NACT + a];
    acc += hv.z * Wfc[(kk * 4 + 2) * NACT + a];
    acc += hv.w * Wfc[(kk * 4 + 3) * NACT + a];
  }
  out[t] = acc;
}

// ---------------------------------------------------------------------------
extern "C" void kernel_launch(void* const* d_in, const int* in_sizes, int n_in,
                              void* d_out, int out_size, void* d_ws,
                              size_t ws_size, hipStream_t stream) {
  const float* x   = (const float*)d_in[0];
  const int*   ei  = (const int*)d_in[1];    // [2, E] (row; col)
  const float* W1  = (const float*)d_in[2];
  const float* b1  = (const float*)d_in[3];
  const float* W2  = (const float*)d_in[4];
  const float* b2  = (const float*)d_in[5];
  const float* Wfc = (const float*)d_in[6];
  const float* bfc = (const float*)d_in[7];
  float* out = (float*)d_out;

  const int N = in_sizes[0] / FDIM;  // 100000
  const int E = in_sizes[1] / 2;     // 1600000

  // Workspace: dinv[N] | bufA[N*64] | bufB[N*64]  (~51.6 MB)
  float* dinv = (float*)d_ws;
  float* bufA = dinv + N;
  float* bufB = bufA + (size_t)N * FDIM;

  const int TB = 256;
  const int nTiles = (N + 15) / 16;
  const int gemmBlocks = (nTiles * 32 + TB - 1) / TB;
  const int nB    = (N + TB - 1) / TB;
  const int eB    = (E + TB - 1) / TB;
  const int n16B  = (N * 16 + TB - 1) / TB;
  const int n64B  = (N * FDIM + TB - 1) / TB;
  const int e16B  = (E * 16 + TB - 1) / TB;
  const int fcB   = (N * NACT + TB - 1) / TB;

  // Normalization: deg = 1 + in-degree; dinv = deg^{-1/2}
  init_deg_kernel<<<nB, TB, 0, stream>>>(dinv, N);
  count_deg_kernel<<<eB, TB, 0, stream>>>(ei + E, dinv, E);
  rsqrt_kernel<<<nB, TB, 0, stream>>>(dinv, N);

  // Layer 1: h = x@W1 ; aggregate ; +b1 ; relu
  gemm64_wmma_kernel<<<gemmBlocks, TB, 0, stream>>>(x, W1, bufA, nTiles);
  self_loop_init_kernel<<<n16B, TB, 0, stream>>>(bufA, dinv, bufB, N);
  scatter_edges_kernel<<<e16B, TB, 0, stream>>>(ei, bufA, dinv, bufB, E);
  bias_relu_kernel<<<n64B, TB, 0, stream>>>(bufB, b1, N * FDIM);

  // Layer 2: h = h1@W2 ; aggregate ; +b2 ; relu   (ping-pong buffers)
  gemm64_wmma_kernel<<<gemmBlocks, TB, 0, stream>>>(bufB, W2, bufA, nTiles);
  self_loop_init_kernel<<<n16B, TB, 0, stream>>>(bufA, dinv, bufB, N);
  scatter_edges_kernel<<<e16B, TB, 0, stream>>>(ei, bufA, dinv, bufB, E);
  bias_relu_kernel<<<n64B, TB, 0, stream>>>(bufB, b2, N * FDIM);

  // FC head
  fc_head_kernel<<<fcB, TB, 0, stream>>>(bufB, Wfc, bfc, out, N);
}